// L_int_9672266351368
// MI455X (gfx1250) — compile-verified
//
#include <hip/hip_runtime.h>

typedef __attribute__((ext_vector_type(2))) float v2f;
typedef __attribute__((ext_vector_type(8))) float v8f;

#define IMG_H   512
#define IMG_W   512
#define BATCH   32
#define TS      16          // output tile side (one wave per tile)
#define HALO    4
#define RDIM    24          // TS + 2*HALO
#define INSTR   25          // input LDS row stride (24 + pad)
#define IN_F    600         // 24*25 floats per input field
#define RCSTR   17          // row-conv LDS row stride (16 + pad)
#define RC_F    408         // 24*17 floats per row-conv field
#define NW      4           // waves per block
#define WAVE_LDS 2448       // max(3*600, 6*408) floats per wave

// Gaussian 9-tap, sigma=1.5, normalized (matches reference _gauss2d 1-D factor)
__constant__ float GWc[9] = {
    0.00761441f, 0.03607497f, 0.10958608f, 0.21344444f, 0.26656005f,
    0.21344444f, 0.10958608f, 0.03607497f, 0.00761441f};

__global__ void zero_out_kernel(float* o) { o[0] = 0.0f; }

__global__ __launch_bounds__(NW * 32)
void fusion_loss_kernel(const float* __restrict__ vis,
                        const float* __restrict__ ir,
                        const float* __restrict__ fus,
                        const float* __restrict__ mask,
                        float* __restrict__ out)
{
    __shared__ float lds[NW][WAVE_LDS];
    __shared__ float gwLds[16];

    const int lane = threadIdx.x & 31;
    const int wave = threadIdx.x >> 5;

    if (threadIdx.x < 9) gwLds[threadIdx.x] = GWc[threadIdx.x];
    __syncthreads();

    // tile id -> (batch, tile-y, tile-x); 32x32 tiles per image, 32 images
    const int tile = blockIdx.x * NW + wave;
    const int b    = tile >> 10;
    const int rem  = tile & 1023;
    const int y0   = (rem >> 5) * TS;
    const int x0   = (rem & 31) * TS;
    const size_t imgBase = (size_t)b * (IMG_H * (size_t)IMG_W);

    float* L = lds[wave];

    // ---------------- Phase A: stage 24x24 haloed tile of vis/ir/fuse -------
    for (int idx = lane; idx < RDIM * RDIM; idx += 32) {
        const int r  = idx / RDIM;
        const int cc = idx - r * RDIM;
        const int yy = y0 - HALO + r;
        const int xx = x0 - HALO + cc;
        const bool inb = (yy >= 0) & (yy < IMG_H) & (xx >= 0) & (xx < IMG_W);
        const size_t g = imgBase + (size_t)yy * IMG_W + xx;
        const float v = inb ? vis[g] : 0.0f;
        const float i = inb ? ir[g]  : 0.0f;
        const float f = inb ? fus[g] : 0.0f;
        L[0 * IN_F + r * INSTR + cc] = v;
        L[1 * IN_F + r * INSTR + cc] = i;
        L[2 * IN_F + r * INSTR + cc] = f;
    }

    // ---------------- Precompute banded A-matrix WMMA fragments -------------
    // A is 16x24: A[m][k] = w[k-m] for 0<=k-m<9 else 0. 16x4 f32 A layout:
    // lanes 0-15 hold M=lane, VGPR0->K=k0, VGPR1->K=k0+1; lanes 16-31 K=k0+2/k0+3.
    const int m    = lane & 15;
    const int koff = (lane >> 4) << 1;            // 0 or 2
    float ab0[6], ab1[6], ag0[6], ag1[6];
#pragma unroll
    for (int kc = 0; kc < 6; ++kc) {
        const int k0 = kc * 4 + koff;
        const int i0 = k0 - m;
        const int i1 = k0 + 1 - m;
        const bool p0 = (i0 >= 0) & (i0 < 9);
        const bool p1 = (i1 >= 0) & (i1 < 9);
        const int c0 = i0 < 0 ? 0 : (i0 > 8 ? 8 : i0);
        const int c1 = i1 < 0 ? 0 : (i1 > 8 ? 8 : i1);
        ab0[kc] = p0 ? 1.0f : 0.0f;
        ab1[kc] = p1 ? 1.0f : 0.0f;
        ag0[kc] = p0 ? gwLds[c0] : 0.0f;
        ag1[kc] = p1 ? gwLds[c1] : 0.0f;
    }

    // ---------------- Phase B: row (x) convolution into registers -----------
    // lane -> column c = lane%16, rows rbase..rbase+11 (two half-waves).
    const float GW[9] = {0.00761441f, 0.03607497f, 0.10958608f, 0.21344444f,
                         0.26656005f, 0.21344444f, 0.10958608f, 0.03607497f,
                         0.00761441f};
    const int c     = lane & 15;
    const int rbase = (lane >> 4) * 12;
    float rc0[12], rc1[12], rc2[12], rc3[12], rc4[12], rc5[12];
#pragma unroll
    for (int rr = 0; rr < 12; ++rr) {
        const int r = rbase + rr;
        float a0 = 0.f, a1 = 0.f, a2 = 0.f, a3 = 0.f, a4 = 0.f, a5 = 0.f;
#pragma unroll
        for (int j = 0; j < 9; ++j) {
            const float v = L[0 * IN_F + r * INSTR + c + j];
            const float i = L[1 * IN_F + r * INSTR + c + j];
            const float f = L[2 * IN_F + r * INSTR + c + j];
            const float di = i - f;
            const float dv = v - f;
            const float w  = GW[j];
            a0 += di * di;          // box( (ir-fuse)^2 ), weight 1
            a1 += dv * dv;          // box( (vis-fuse)^2 )
            a2 += w * i;            // gauss( ir )
            a3 += w * i * i;        // gauss( ir^2 )
            a4 += w * v;            // gauss( vis )
            a5 += w * v * v;        // gauss( vis^2 )
        }
        rc0[rr] = a0; rc1[rr] = a1; rc2[rr] = a2;
        rc3[rr] = a3; rc4[rr] = a4; rc5[rr] = a5;
    }
    // Re-write the wave's LDS slice in row-conv (B-matrix) layout.
    // Safe: per-wave LDS ops are in-order; all reads above precede these writes.
#pragma unroll
    for (int rr = 0; rr < 12; ++rr) {
        const int r = rbase + rr;
        L[0 * RC_F + r * RCSTR + c] = rc0[rr];
        L[1 * RC_F + r * RCSTR + c] = rc1[rr];
        L[2 * RC_F + r * RCSTR + c] = rc2[rr];
        L[3 * RC_F + r * RCSTR + c] = rc3[rr];
        L[4 * RC_F + r * RCSTR + c] = rc4[rr];
        L[5 * RC_F + r * RCSTR + c] = rc5[rr];
    }

    // ---------------- Phase C: column (y) conv via V_WMMA_F32_16X16X4_F32 ---
    // out(16x16) = A(16x24 band) x B(24x16), K accumulated in 6 chunks of 4.
    // B 4x16 layout: VGPR0 -> K=ks (lanes 0-15) / K=ks+2 (lanes 16-31); VGPR1 -> +1.
    v8f C0 = {}, C1 = {}, C2 = {}, C3 = {}, C4 = {}, C5 = {};
#pragma unroll
    for (int kc = 0; kc < 6; ++kc) {
        const int ks = kc * 4 + koff;
        const v2f Abox = {ab0[kc], ab1[kc]};
        const v2f Agau = {ag0[kc], ag1[kc]};
        v2f B;
        B.x = L[0 * RC_F + ks * RCSTR + c];
        B.y = L[0 * RC_F + (ks + 1) * RCSTR + c];
        C0 = __builtin_amdgcn_wmma_f32_16x16x4_f32(false, Abox, false, B, (short)0, C0, false, false);
        B.x = L[1 * RC_F + ks * RCSTR + c];
        B.y = L[1 * RC_F + (ks + 1) * RCSTR + c];
        C1 = __builtin_amdgcn_wmma_f32_16x16x4_f32(false, Abox, false, B, (short)0, C1, false, false);
        B.x = L[2 * RC_F + ks * RCSTR + c];
        B.y = L[2 * RC_F + (ks + 1) * RCSTR + c];
        C2 = __builtin_amdgcn_wmma_f32_16x16x4_f32(false, Agau, false, B, (short)0, C2, false, false);
        B.x = L[3 * RC_F + ks * RCSTR + c];
        B.y = L[3 * RC_F + (ks + 1) * RCSTR + c];
        C3 = __builtin_amdgcn_wmma_f32_16x16x4_f32(false, Agau, false, B, (short)0, C3, false, false);
        B.x = L[4 * RC_F + ks * RCSTR + c];
        B.y = L[4 * RC_F + (ks + 1) * RCSTR + c];
        C4 = __builtin_amdgcn_wmma_f32_16x16x4_f32(false, Agau, false, B, (short)0, C4, false, false);
        B.x = L[5 * RC_F + ks * RCSTR + c];
        B.y = L[5 * RC_F + (ks + 1) * RCSTR + c];
        C5 = __builtin_amdgcn_wmma_f32_16x16x4_f32(false, Agau, false, B, (short)0, C5, false, false);
    }

    // ---------------- Phase D: select maps, reduce to scalar ----------------
    // C/D layout: VGPR v -> image row y0 + v + 8*(lane>=16), column x0 + lane%16.
    float lsum = 0.0f;
    const int mhi = (lane >> 4) << 3;
#pragma unroll
    for (int v = 0; v < 8; ++v) {
        const int M = v + mhi;
        const float mse_ir = C0[v] * (1.0f / 81.0f);
        const float mse_vi = C1[v] * (1.0f / 81.0f);
        const float var_ir = C3[v] - C2[v] * C2[v];
        const float var_vi = C5[v] - C4[v] * C4[v];
        const float mk = mask[imgBase + (size_t)(y0 + M) * IMG_W + (x0 + c)];
        const float map1   = (var_ir - var_vi > 0.0f) ? 1.0f : 0.0f;
        const float map_ir = (map1 + mk > 0.0f) ? 1.0f : 0.0f;
        lsum += map_ir * mse_ir + (1.0f - map_ir) * mse_vi;
    }
#pragma unroll
    for (int off = 16; off > 0; off >>= 1)
        lsum += __shfl_down(lsum, off, 32);
    if (lane == 0)
        atomicAdd(out, lsum * (1.0f / ((float)BATCH * IMG_H * IMG_W)));
}

extern "C" void kernel_launch(void* const* d_in, const int* in_sizes, int n_in,
                              void* d_out, int out_size, void* d_ws, size_t ws_size,
                              hipStream_t stream) {
    (void)in_sizes; (void)n_in; (void)out_size; (void)d_ws; (void)ws_size;
    const float* vis  = (const float*)d_in[0];
    const float* ir   = (const float*)d_in[1];
    const float* fus  = (const float*)d_in[2];
    const float* mask = (const float*)d_in[3];
    float* out = (float*)d_out;

    zero_out_kernel<<<1, 1, 0, stream>>>(out);

    const int tiles  = BATCH * (IMG_H / TS) * (IMG_W / TS);  // 32768
    const int blocks = tiles / NW;                           // 8192
    fusion_loss_kernel<<<blocks, NW * 32, 0, stream>>>(vis, ir, fus, mask, out);
}